// RandomCut_3702261809408
// MI455X (gfx1250) — compile-verified
//
#include <hip/hip_runtime.h>

// CDNA5 / gfx1250. Memory-bound skinny GEMM (32x8192 @ 8192x8192) on the
// f32 WMMA path, plus gather/scatter epilogue.
//
// Roofline: 2*32*8192*8192 = 4.3 GFLOP vs 256 MB of anc_matrix streamed once
// -> ~11 us floor at 23.3 TB/s. f32 WMMA avoids any conversion work (data are
// exact 0/1 floats, anc is a small exact integer count).

typedef __attribute__((ext_vector_type(2))) float v2f;
typedef __attribute__((ext_vector_type(8))) float v8f;

#define NN 8192
#define BB 32
#define KSPLIT 16                 // K split: (NN/32) * 16 = 4096 waves
#define KCHUNK (NN / KSPLIT)      // 512

// ---------------------------------------------------------------- zero ws
__global__ void zero_ws_kernel(unsigned* __restrict__ p, int n) {
    int i = blockIdx.x * blockDim.x + threadIdx.x;
    if (i < n) p[i] = 0u;
}

// ------------------------------------------------------------- anc = drop @ M
// One wave computes a 32(M) x 32(N) tile over a 512-long K chunk using
// V_WMMA_F32_16X16X4_F32 (4 accumulators), then accumulates integer counts
// with native global_atomic_add_u32.
//
// WMMA f32 16x16x4 operand layout (lane l, VGPR v):
//   A (16x4): m = l%16, k = 2*(l>=16) + v   -> k,k+1 contiguous => b64 load
//   B (4x16): n = l%16, k = 2*(l>=16) + v   -> two b32 loads (row stride NN)
//   C (16x16): vgpr r, lanes0-15: M=r, N=l%16 ; lanes16-31: M=r+8
//
// A 32-wide N tile means each B row access consumes exactly one 128B line.
#define WMMA_F32(A, B, C) \
    __builtin_amdgcn_wmma_f32_16x16x4_f32(false, (A), false, (B), (short)0, (C), false, false)

__global__ __launch_bounds__(256) void anc_gemm_kernel(
        const float* __restrict__ drop,
        const float* __restrict__ M,
        unsigned* __restrict__ anc) {
    const int lane  = threadIdx.x & 31;
    // Wave-uniform tile id in an SGPR -> scalar k-loop, EXEC untouched.
    const int wid   = __builtin_amdgcn_readfirstlane(
                          (blockIdx.x * blockDim.x + threadIdx.x) >> 5);
    const int npair = wid & (NN / 32 - 1);       // 0..255
    const int kch   = wid / (NN / 32);           // 0..KSPLIT-1
    const int n0    = npair << 5;                // 32 columns per wave
    const int k0    = kch * KCHUNK;
    const int lhalf = lane >> 4;                 // 0 | 1
    const int lmod  = lane & 15;
    const int koff  = lhalf << 1;                // per-lane K sub-offset

    const float* A0 = drop + (size_t)lmod * NN;          // batch rows 0..15
    const float* A1 = drop + (size_t)(lmod + 16) * NN;   // batch rows 16..31
    const float* B0 = M + n0 + lmod;                     // cols n0   .. n0+15
    const float* B1 = B0 + 16;                           // cols n0+16.. n0+31

    v8f c00 = {}, c01 = {}, c10 = {}, c11 = {};

    int k = k0;
    if (k0 == 0) {
        // Peeled first step (uniform branch): apply drop[:,0] = 0 here so the
        // steady-state loop carries no per-iteration selects.
        const int ka = koff;
        v2f a0 = *(const v2f*)(A0 + ka);
        v2f a1 = *(const v2f*)(A1 + ka);
        if (ka == 0) { a0.x = 0.0f; a1.x = 0.0f; }
        v2f b0, b1;
        b0.x = B0[(size_t)ka * NN];  b0.y = B0[(size_t)(ka + 1) * NN];
        b1.x = B1[(size_t)ka * NN];  b1.y = B1[(size_t)(ka + 1) * NN];
        c00 = WMMA_F32(a0, b0, c00);
        c01 = WMMA_F32(a1, b0, c01);
        c10 = WMMA_F32(a0, b1, c10);
        c11 = WMMA_F32(a1, b1, c11);
        k = 4;
    }

    for (; k < k0 + KCHUNK; k += 4) {
        const int ka = k + koff;
        v2f a0 = *(const v2f*)(A0 + ka);
        v2f a1 = *(const v2f*)(A1 + ka);
        v2f b0, b1;
        b0.x = B0[(size_t)ka * NN];  b0.y = B0[(size_t)(ka + 1) * NN];
        b1.x = B1[(size_t)ka * NN];  b1.y = B1[(size_t)(ka + 1) * NN];

        // Stream-ahead prefetch of both row parities (global_prefetch_b8);
        // each B row is exactly one 128B line for this wave's 32 columns.
        int kp = ka + 16; kp = kp < (NN - 2) ? kp : (NN - 2);
        __builtin_prefetch(B0 + (size_t)kp * NN, 0, 1);
        __builtin_prefetch(B0 + (size_t)(kp + 1) * NN, 0, 1);

        c00 = WMMA_F32(a0, b0, c00);
        c01 = WMMA_F32(a1, b0, c01);
        c10 = WMMA_F32(a0, b1, c10);
        c11 = WMMA_F32(a1, b1, c11);
    }

    const int col0 = n0 + lmod;
#pragma unroll
    for (int r = 0; r < 8; ++r) {
        int row = r + (lhalf << 3);
        unsigned u00 = (unsigned)(c00[r] + 0.5f);
        unsigned u01 = (unsigned)(c01[r] + 0.5f);
        unsigned u10 = (unsigned)(c10[r] + 0.5f);
        unsigned u11 = (unsigned)(c11[r] + 0.5f);
        if (u00) atomicAdd(&anc[(size_t)row * NN + col0], u00);
        if (u01) atomicAdd(&anc[(size_t)(row + 16) * NN + col0], u01);
        if (u10) atomicAdd(&anc[(size_t)row * NN + col0 + 16], u10);
        if (u11) atomicAdd(&anc[(size_t)(row + 16) * NN + col0 + 16], u11);
    }
}

// ----------------------------------------------- gather parent + child counts
// subtree[b,i] = (i==0) ? 1 : (anc[b,parent[i]] == 0)
// num_children[b,parent[i]] += subtree[b,i]   (i >= 1)
__global__ void gather_scatter_kernel(const unsigned* __restrict__ anc,
                                      const int* __restrict__ parent,
                                      unsigned* __restrict__ nc,
                                      float* __restrict__ out) {
    int idx = blockIdx.x * blockDim.x + threadIdx.x;
    if (idx >= BB * NN) return;
    int b = idx >> 13;            // / NN
    int i = idx & (NN - 1);
    unsigned t;
    if (i == 0) {
        t = 1u;
    } else {
        int p = parent[i];
        t = (anc[(size_t)b * NN + p] == 0u) ? 1u : 0u;
        if (t) atomicAdd(&nc[(size_t)b * NN + p], 1u);
    }
    out[idx] = (float)t;          // park subtree in d_out
}

// ------------------------------------------------------------------ boundary
__global__ void boundary_kernel(const unsigned* __restrict__ nc,
                                float* __restrict__ out) {
    int idx = blockIdx.x * blockDim.x + threadIdx.x;
    if (idx >= BB * NN) return;
    float s = out[idx];
    out[idx] = (s != 0.0f && nc[idx] == 0u) ? 1.0f : 0.0f;
}

// ------------------------------------------------------------------- launch
extern "C" void kernel_launch(void* const* d_in, const int* in_sizes, int n_in,
                              void* d_out, int out_size, void* d_ws, size_t ws_size,
                              hipStream_t stream) {
    const float* drop   = (const float*)d_in[0];   // (32, 8192) f32 of 0/1
    const float* M      = (const float*)d_in[1];   // (8192, 8192) f32 of 0/1
    const int*   parent = (const int*)d_in[2];     // (8192,) i32

    unsigned* anc = (unsigned*)d_ws;               // BB*NN counts
    unsigned* nc  = anc + (size_t)BB * NN;         // BB*NN child counts
    float*    out = (float*)d_out;                 // (32, 8192) f32 bool

    const int ztotal = 2 * BB * NN;
    zero_ws_kernel<<<(ztotal + 255) / 256, 256, 0, stream>>>(anc, ztotal);

    const int waves  = (NN / 32) * KSPLIT;         // 4096 waves
    const int blocks = waves / 8;                  // 256 thr = 8 waves/block
    anc_gemm_kernel<<<blocks, 256, 0, stream>>>(drop, M, anc);

    const int total = BB * NN;
    gather_scatter_kernel<<<(total + 255) / 256, 256, 0, stream>>>(anc, parent, nc, out);
    boundary_kernel<<<(total + 255) / 256, 256, 0, stream>>>(nc, out);
}